// FullQuantumModel_20134806683739
// MI455X (gfx1250) — compile-verified
//
#include <hip/hip_runtime.h>

// 14-qubit, 8-layer parameterized quantum circuit, BATCH=32.
// One workgroup (256 threads = 8 wave32) per batch element; full state
// double-buffered in LDS (256 KB of the 320 KB/WGP). Single-qubit layers are
// fused into 16x16 group unitaries and applied with v_wmma_f32_16x16x4_f32.
// The CNOT ring is a closed-form index permutation folded into load addresses.

#define NQ       14
#define NL       8
#define QC_BATCH 32
#define NSTATE   (1 << NQ)          // 16384 amplitudes
#define THREADS  256
#define WAVES    (THREADS / 32)     // 8 wave32
#define NCOLS    (NSTATE / 16)      // 1024 columns per group-matmul view
#define NTILES   (NCOLS / 16)       // 64 column tiles of 16
#define TPW      (NTILES / WAVES)   // 8 tiles per wave per stage

// Dynamic LDS layout (in floats)
#define OFF_RE   0                        // float[2][NSTATE]  (ping/pong real)
#define OFF_IM   (2 * NSTATE)             // float[2][NSTATE]  (ping/pong imag)
#define OFF_U2   (4 * NSTATE)             // float[NL*NQ*8]    (2x2 unitaries)
#define OFF_U16R (OFF_U2 + NL * NQ * 8)   // float[256]        (16x16 group U, re)
#define OFF_U16I (OFF_U16R + 256)         // float[256]        (16x16 group U, im)
#define SMEM_FLOATS (OFF_U16I + 256)
#define SMEM_BYTES  (SMEM_FLOATS * 4)     // 267,776 bytes  (< 320 KB/WGP)

typedef __attribute__((ext_vector_type(2))) float v2f;
typedef __attribute__((ext_vector_type(8))) float v8f;

struct cpx { float re, im; };
__device__ __forceinline__ cpx cmul(cpx a, cpx b) {
    cpx r; r.re = a.re * b.re - a.im * b.im; r.im = a.re * b.im + a.im * b.re; return r;
}

// Permutation of the full CNOT ring CNOT(q,(q+1)%14), q = 0..13.
// Qubit q lives at bit (13-q). psi_out[i] = psi_in[sigma(i)], with
// sigma(j): bits k=0..11 -> b_k ^ b_{k+1}; bit12 -> b12^b13^b0; bit13 -> b13^b0.
__device__ __forceinline__ int cnot_ring_perm(int j) {
    return j ^ ((j >> 1) & 0x1FFF) ^ ((j & 1) ? 0x3000 : 0);
}

__global__ __launch_bounds__(THREADS, 1)
void qc_full_model_kernel(const float* __restrict__ state,
                          const float* __restrict__ params,
                          float* __restrict__ out)
{
    extern __shared__ float smem[];
    float* sRe  = smem + OFF_RE;    // [buf * NSTATE + idx]
    float* sIm  = smem + OFF_IM;
    float* sU2  = smem + OFF_U2;    // [(l*NQ+q)*8 + (rb*2+sb)*2 + {re,im}]
    float* u16r = smem + OFF_U16R;  // [r*16 + s]
    float* u16i = smem + OFF_U16I;

    const int tid  = threadIdx.x;
    const int lane = tid & 31;
    const int wave = tid >> 5;
    const int b    = blockIdx.x;

    // ---- Load real input state into buffer 0; imag = 0 (coalesced b128 loads)
    const float4* gs = (const float4*)(state + (size_t)b * NSTATE);
    #pragma unroll
    for (int c = 0; c < NSTATE / 4 / THREADS; ++c) {   // 16 iterations
        int i4 = c * THREADS + tid;
        float4 v = gs[i4];
        int i = i4 * 4;
        sRe[i + 0] = v.x; sRe[i + 1] = v.y; sRe[i + 2] = v.z; sRe[i + 3] = v.w;
        sIm[i + 0] = 0.f; sIm[i + 1] = 0.f; sIm[i + 2] = 0.f; sIm[i + 3] = 0.f;
    }

    // ---- Build all 112 single-qubit unitaries U = RZ(t2) RY(t1) RX(t0)
    if (tid < NL * NQ) {
        const float* p3 = params + tid * 3;
        float hx = 0.5f * p3[0], hy = 0.5f * p3[1], hz = 0.5f * p3[2];
        float cx = cosf(hx), sx = sinf(hx);
        float cy = cosf(hy), sy = sinf(hy);
        float cz = cosf(hz), sz = sinf(hz);
        // M = RY @ RX
        cpx m00 = {  cy * cx,  sy * sx };
        cpx m01 = { -sy * cx, -cy * sx };
        cpx m10 = {  sy * cx, -cy * sx };
        cpx m11 = {  cy * cx, -sy * sx };
        cpx ezm = { cz, -sz };   // e^{-i tz/2}  (row 0 of RZ)
        cpx ezp = { cz,  sz };   // e^{+i tz/2}  (row 1 of RZ)
        cpx u00 = cmul(ezm, m00), u01 = cmul(ezm, m01);
        cpx u10 = cmul(ezp, m10), u11 = cmul(ezp, m11);
        float* d = sU2 + tid * 8;
        d[0] = u00.re; d[1] = u00.im; d[2] = u01.re; d[3] = u01.im;
        d[4] = u10.re; d[5] = u10.im; d[6] = u11.re; d[7] = u11.im;
    }
    __syncthreads();

    int phase = 0;  // current source buffer
    #pragma unroll 1
    for (int l = 0; l < NL; ++l) {
        #pragma unroll 1
        for (int g = 0; g < 4; ++g) {
            // ---- Build 16x16 group unitary = kron of four 2x2 factors.
            // g0: qubits 0-3 (bits 13..10), g1: 4-7 (9..6), g2: 8-11 (5..2),
            // g3: I,I,U12,U13 on bits 3..0.
            {
                int r = tid >> 4, s = tid & 15;
                cpx acc = { 1.f, 0.f };
                #pragma unroll
                for (int f = 0; f < 4; ++f) {
                    int rb = (r >> (3 - f)) & 1;
                    int sb = (s >> (3 - f)) & 1;
                    int qf = (g == 3) ? ((f < 2) ? -1 : (10 + f)) : (g * 4 + f);
                    if (qf < 0) {
                        if (rb != sb) { acc.re = 0.f; acc.im = 0.f; }
                    } else {
                        const float* e = sU2 + ((l * NQ + qf) * 8 + (rb * 2 + sb) * 2);
                        cpx u; u.re = e[0]; u.im = e[1];
                        acc = cmul(acc, u);
                    }
                }
                u16r[tid] = acc.re;
                u16i[tid] = acc.im;
            }
            __syncthreads();

            const int p      = (g == 0) ? 10 : (g == 1) ? 6 : (g == 2) ? 2 : 0;
            const int lomask = (1 << p) - 1;
            const int n      = lane & 15;      // A: row m; B/D: column n
            const int half   = lane >> 4;      // 0 or 1
            const int koff   = half * 2;       // A/B second-half K offset
            const bool doPerm = (g == 0) && (l > 0);  // fold previous layer's CNOT ring

            // ---- A operands: U16 chunks in WMMA A-layout (m=lane&15, K=4k+v+koff)
            v2f aR[4], aI[4], aIn[4];
            #pragma unroll
            for (int k = 0; k < 4; ++k) {
                int rb = n * 16 + 4 * k + koff;
                aR[k][0] = u16r[rb + 0];  aR[k][1] = u16r[rb + 1];
                aI[k][0] = u16i[rb + 0];  aI[k][1] = u16i[rb + 1];
                aIn[k][0] = -aI[k][0];    aIn[k][1] = -aI[k][1];   // f32 WMMA has no A-neg modifier
            }

            const int src = phase * NSTATE, dst = (phase ^ 1) * NSTATE;

            #pragma unroll 1
            for (int it = 0; it < TPW; ++it) {
                int t = wave * TPW + it;
                int c = t * 16 + n;
                int base = ((c & ~lomask) << 4) | (c & lomask);  // group bits zeroed, hole at p..p+3

                // B operands: state columns in WMMA B-layout (n=lane&15, K=4k+v+koff)
                v2f bR[4], bI[4];
                #pragma unroll
                for (int k = 0; k < 4; ++k) {
                    #pragma unroll
                    for (int v = 0; v < 2; ++v) {
                        int K = 4 * k + v + koff;
                        int idx = base + (K << p);
                        if (doPerm) idx = cnot_ring_perm(idx);
                        bR[k][v] = sRe[src + idx];
                        bI[k][v] = sIm[src + idx];
                    }
                }

                v8f dR = { 0.f, 0.f, 0.f, 0.f, 0.f, 0.f, 0.f, 0.f };
                v8f dI = { 0.f, 0.f, 0.f, 0.f, 0.f, 0.f, 0.f, 0.f };
                #pragma unroll
                for (int k = 0; k < 4; ++k) {
                    // Dr = Ur*Sr - Ui*Si ; Di = Ur*Si + Ui*Sr
                    dR = __builtin_amdgcn_wmma_f32_16x16x4_f32(false, aR[k],  false, bR[k], (short)0, dR, false, false);
                    dR = __builtin_amdgcn_wmma_f32_16x16x4_f32(false, aIn[k], false, bI[k], (short)0, dR, false, false);
                    dI = __builtin_amdgcn_wmma_f32_16x16x4_f32(false, aR[k],  false, bI[k], (short)0, dI, false, false);
                    dI = __builtin_amdgcn_wmma_f32_16x16x4_f32(false, aI[k],  false, bR[k], (short)0, dI, false, false);
                }

                // D store: C/D layout (m = v + 8*half, n = lane&15)
                #pragma unroll
                for (int v = 0; v < 8; ++v) {
                    int m = v + half * 8;
                    int idx = base + (m << p);
                    sRe[dst + idx] = dR[v];
                    sIm[dst + idx] = dI[v];
                }
            }
            phase ^= 1;
            __syncthreads();
        }
    }

    // ---- P(qubit0 = |0>) = sum over first half, through the final CNOT ring
    // (last layer's ring folded here via sigma; |.|^2 only needs the gather).
    const int fin = phase * NSTATE;  // phase == 0 after 32 ping-pongs
    float acc = 0.f;
    #pragma unroll 1
    for (int k = 0; k < (NSTATE / 2) / THREADS; ++k) {   // 32 per thread
        int i = tid * ((NSTATE / 2) / THREADS) + k;
        int j = cnot_ring_perm(i);
        float re = sRe[fin + j], im = sIm[fin + j];
        acc += re * re + im * im;
    }
    u16r[tid] = acc;   // reuse staging (safe: barrier above)
    __syncthreads();
    if (tid == 0) {
        float s = 0.f;
        #pragma unroll 1
        for (int k = 0; k < THREADS; ++k) s += u16r[k];
        out[b] = s;
    }
}

extern "C" void kernel_launch(void* const* d_in, const int* in_sizes, int n_in,
                              void* d_out, int out_size, void* d_ws, size_t ws_size,
                              hipStream_t stream) {
    (void)in_sizes; (void)n_in; (void)out_size; (void)d_ws; (void)ws_size;
    const float* state  = (const float*)d_in[0];   // (32, 16384) f32
    const float* params = (const float*)d_in[1];   // (8, 14, 3)  f32
    float* out = (float*)d_out;                    // (32,)       f32
    qc_full_model_kernel<<<QC_BATCH, THREADS, SMEM_BYTES, stream>>>(state, params, out);
}